// LSTM_net_29678224015481
// MI455X (gfx1250) — compile-verified
//
#include <hip/hip_runtime.h>

// ---------------------------------------------------------------------------
// MI455X (gfx1250) implementation.
//  Kernel A (head_compose): folds conv1+conv2+deconv1+deconv2+fc1+fc2 into a
//      single linear map M[16x64] + bias[16]  (head is linear between the
//      sigmoid and the final tanh).  Written to d_ws.
//  Kernel B (lstm_wmma):  one wave == 16 batch rows.  All LSTM gate GEMMs and
//      the fused head GEMM run on V_WMMA_F32_16X16X4_F32 (full f32 precision).
//      c0 state tiles are fetched with GLOBAL_LOAD_ASYNC_TO_LDS_B128.
// ---------------------------------------------------------------------------

typedef float v2f __attribute__((ext_vector_type(2)));
typedef float v4f __attribute__((ext_vector_type(4)));
typedef float v8f __attribute__((ext_vector_type(8)));

#define BT    131072   // batch
#define WAVES 4        // waves per block (static LDS ~90.7 KB of 320 KB/WGP)
#define TPB   (WAVES*32)

__device__ __forceinline__ v8f wmma4(v2f a, v2f b, v8f c) {
  // D = A(16x4 f32) * B(4x16 f32) + C(16x16 f32)
  return __builtin_amdgcn_wmma_f32_16x16x4_f32(false, a, false, b, (short)0, c,
                                               false, false);
}
// fast sigmoid/tanh: v_exp_f32 + v_rcp_f32 (no IEEE div sequences)
__device__ __forceinline__ float sigm(float x) {
  return __builtin_amdgcn_rcpf(1.0f + __expf(-x));
}
__device__ __forceinline__ float tanh_f(float x) {
  float e = __expf(-2.0f * fabsf(x));
  float r = (1.0f - e) * __builtin_amdgcn_rcpf(1.0f + e);
  return copysignf(r, x);
}

// ---------------------------------------------------------------------------
// Kernel A: compose the linear head into M (16x64, row n / col k) + bias(16).
// Column j of M = head(e_j) with zero biases; col 64 computes the bias vector.
// ---------------------------------------------------------------------------
__global__ void head_compose_kernel(
    const float* __restrict__ c1w, const float* __restrict__ c1b,
    const float* __restrict__ c2w, const float* __restrict__ c2b,
    const float* __restrict__ d1w, const float* __restrict__ d1b,
    const float* __restrict__ d2w, const float* __restrict__ d2b,
    const float* __restrict__ f1w, const float* __restrict__ f1b,
    const float* __restrict__ f2w, const float* __restrict__ f2b,
    float* __restrict__ Mws) {
  int col = threadIdx.x;
  if (col > 64) return;
  float bsc = (col == 64) ? 1.0f : 0.0f;
  float s[64];
#pragma clang loop unroll(disable)
  for (int i = 0; i < 64; ++i) s[i] = (i == col) ? 1.0f : 0.0f;

  float t1[8][15];                                 // conv1: [8,4,2]
#pragma clang loop unroll(disable)
  for (int o = 0; o < 8; ++o)
#pragma clang loop unroll(disable)
    for (int p = 0; p < 15; ++p) {
      float a = bsc * c1b[o];
      for (int i = 0; i < 4; ++i)
        for (int k = 0; k < 2; ++k) a += s[i * 16 + p + k] * c1w[(o * 4 + i) * 2 + k];
      t1[o][p] = a;
    }
  float t2[16][14];                                // conv2: [16,8,2]
#pragma clang loop unroll(disable)
  for (int o = 0; o < 16; ++o)
#pragma clang loop unroll(disable)
    for (int p = 0; p < 14; ++p) {
      float a = bsc * c2b[o];
      for (int i = 0; i < 8; ++i)
        for (int k = 0; k < 2; ++k) a += t1[i][p + k] * c2w[(o * 8 + i) * 2 + k];
      t2[o][p] = a;
    }
  float t3[8][15];                                 // deconv1: w [16,8,2]
#pragma clang loop unroll(disable)
  for (int o = 0; o < 8; ++o)
#pragma clang loop unroll(disable)
    for (int p = 0; p < 15; ++p) {
      float a = bsc * d1b[o];
      for (int i = 0; i < 16; ++i)
        for (int k = 0; k < 2; ++k) {
          int q = p - k;
          if (q >= 0 && q < 14) a += t2[i][q] * d1w[(i * 8 + o) * 2 + k];
        }
      t3[o][p] = a;
    }
  float t4[4][16];                                 // deconv2: w [8,4,2]
#pragma clang loop unroll(disable)
  for (int c = 0; c < 4; ++c)
#pragma clang loop unroll(disable)
    for (int p = 0; p < 16; ++p) {
      float a = bsc * d2b[c];
      for (int i = 0; i < 8; ++i)
        for (int k = 0; k < 2; ++k) {
          int q = p - k;
          if (q >= 0 && q < 15) a += t3[i][q] * d2w[(i * 4 + c) * 2 + k];
        }
      t4[c][p] = a;
    }
  float t5[4][8];                                  // fc1: [8,16]
#pragma clang loop unroll(disable)
  for (int c = 0; c < 4; ++c)
#pragma clang loop unroll(disable)
    for (int j = 0; j < 8; ++j) {
      float a = bsc * f1b[j];
      for (int p = 0; p < 16; ++p) a += t4[c][p] * f1w[j * 16 + p];
      t5[c][j] = a;
    }
#pragma clang loop unroll(disable)
  for (int c = 0; c < 4; ++c)                      // fc2: [4,8]
#pragma clang loop unroll(disable)
    for (int q = 0; q < 4; ++q) {
      float a = bsc * f2b[q];
      for (int j = 0; j < 8; ++j) a += t5[c][j] * f2w[q * 8 + j];
      int n = c * 4 + q;
      if (col < 64) Mws[n * 64 + col] = a;
      else          Mws[1024 + n]     = a;
    }
}

// ---------------------------------------------------------------------------
// One LSTM direction over T=4 steps for a 16-batch tile (one wave).
// KIN = input feature count (2 or 16), NKT = K-tiles of 4 (3 or 6).
// A-combined: [y(KIN) | h(8) | pad] ; W is n-major [32][K] in LDS.
// Gate order i,f,g,o == N-halves {0..15},{16..31}.
// ---------------------------------------------------------------------------
template <int KIN, int NKT>
__device__ __forceinline__ void lstm_dir(
    float (*ain)[16][24], float (*yout)[16][16], float (*cst)[8],
    float (*gat)[16], const float* __restrict__ W,
    const float* __restrict__ bvec, const float* __restrict__ h0p,
    const float* __restrict__ c0p, int d, int ln, int lo, int hi) {
  constexpr int K = NKT * 4;

  // ---- async fetch of c0 tile straight into LDS cell-state ---------------
  // Each lane copies 16 contiguous bytes; LDS dest (cst) is 16B aligned and
  // lane-contiguous.  Prior direction's ds_stores to cst use DScnt, async
  // uses ASYNCcnt -> fence DScnt first, wait ASYNCcnt before first use.
  {
    asm volatile("s_wait_dscnt 0x0" ::: "memory");
    unsigned ldsdst = (unsigned)(uintptr_t)(&cst[0][0]) + (unsigned)ln * 16u;
    unsigned voff = (unsigned)ln * 16u;
    unsigned long long gaddr = (unsigned long long)(uintptr_t)c0p;
    asm volatile("global_load_async_to_lds_b128 %0, %1, %2"
                 :: "v"(ldsdst), "v"(voff), "s"(gaddr)
                 : "memory");
  }

  // B fragments (weights): lane -> (k = 4*kt + 2*hi + v, n = 16*nh + lo)
  v2f bf[NKT][2];
#pragma unroll
  for (int kt = 0; kt < NKT; ++kt)
#pragma unroll
    for (int nh = 0; nh < 2; ++nh)
      bf[kt][nh] = *(const v2f*)&W[(nh * 16 + lo) * K + kt * 4 + hi * 2];
  float bias0 = bvec[lo];
  float bias1 = bvec[16 + lo];

  const int tf = d ? 3 : 0, step = d ? -1 : 1;
  {  // stage h0 -> ain[tf][m][KIN..KIN+7]  (coalesced b128 load)
    v4f hv = *(const v4f*)(h0p + ln * 4);
    int m = ln >> 1, k0 = (ln & 1) * 4;
    v2f ha = {hv[0], hv[1]}, hb = {hv[2], hv[3]};
    *(v2f*)&ain[tf][m][KIN + k0]     = ha;
    *(v2f*)&ain[tf][m][KIN + k0 + 2] = hb;
  }
  asm volatile("s_wait_asynccnt 0x0" ::: "memory");
  __builtin_amdgcn_wave_barrier();

#pragma unroll
  for (int s = 0; s < 4; ++s) {
    int t = tf + s * step;
    // A fragments: lane -> (m = lo, k = 4*kt + 2*hi + v)
    v2f af[NKT];
#pragma unroll
    for (int kt = 0; kt < NKT; ++kt)
      af[kt] = *(const v2f*)&ain[t][lo][kt * 4 + hi * 2];

    v8f a0 = {bias0, bias0, bias0, bias0, bias0, bias0, bias0, bias0};
    v8f a1 = {bias1, bias1, bias1, bias1, bias1, bias1, bias1, bias1};
#pragma unroll
    for (int kt = 0; kt < NKT; ++kt) {
      a0 = wmma4(af[kt], bf[kt][0], a0);   // gates i,f (n = 0..15)
      a1 = wmma4(af[kt], bf[kt][1], a1);   // gates g,o (n = 16..31)
    }
    // D layout: (m = r + 8*hi, n = lo).  Store n-major so rows are contiguous.
    v4f p0 = {a0[0], a0[1], a0[2], a0[3]}, p1 = {a0[4], a0[5], a0[6], a0[7]};
    v4f q0 = {a1[0], a1[1], a1[2], a1[3]}, q1 = {a1[4], a1[5], a1[6], a1[7]};
    *(v4f*)&gat[lo][hi * 8]          = p0;
    *(v4f*)&gat[lo][hi * 8 + 4]      = p1;
    *(v4f*)&gat[16 + lo][hi * 8]     = q0;
    *(v4f*)&gat[16 + lo][hi * 8 + 4] = q1;
    __builtin_amdgcn_wave_barrier();

    // elementwise cell: lane covers 4 (m,j) pairs of the 16x8 state tile
#pragma unroll
    for (int p = 0; p < 4; ++p) {
      int flat = p * 32 + ln;
      int m = flat >> 3, j = flat & 7;
      float cn = sigm(gat[j + 8][m]) * cst[m][j] +
                 sigm(gat[j][m]) * tanh_f(gat[j + 16][m]);
      float hn = sigm(gat[j + 24][m]) * tanh_f(cn);
      cst[m][j] = cn;
      yout[t][m][d * 8 + j] = hn;
      if (s < 3) ain[t + step][m][KIN + j] = hn;
    }
    __builtin_amdgcn_wave_barrier();
  }
}

// ---------------------------------------------------------------------------
// Kernel B: full network per 16-batch tile.
// ---------------------------------------------------------------------------
__global__ void __launch_bounds__(TPB) lstm_wmma_kernel(
    const float* __restrict__ x,     const float* __restrict__ h0,
    const float* __restrict__ c0,    const float* __restrict__ w_ih0,
    const float* __restrict__ w_hh0, const float* __restrict__ b_ih0,
    const float* __restrict__ b_hh0, const float* __restrict__ w_ih,
    const float* __restrict__ w_hh,  const float* __restrict__ b_ih,
    const float* __restrict__ b_hh,  const float* __restrict__ Mws,
    float* __restrict__ out) {
  __shared__ float sW[8448];            // weights, n-major [32][K] per (l,d)
  __shared__ float sB[384];             // b_ih + b_hh per (l,d)
  __shared__ float sM[1024];            // head M  [n=16][k=64]
  __shared__ float sMb[16];             // head bias
  __shared__ float sAin[WAVES][4][16][24];   // [t][m][feat|h|pad]
  __shared__ float sYout[WAVES][4][16][16];  // layer output [t][m][16]
  __shared__ float sC[WAVES][16][8];         // cell state
  __shared__ float sG[WAVES][32][16];        // gate tile, n-major

  const int tid = threadIdx.x;
  // ---- block-wide weight staging into WMMA-B-friendly layout -------------
  for (int l = 0; l < 6; ++l) {
    const int K = l ? 24 : 12, Kin = l ? 16 : 2;
    for (int d = 0; d < 2; ++d) {
      float* W = &sW[l ? (768 + ((l - 1) * 2 + d) * 768) : d * 384];
      for (int i = tid; i < 32 * K; i += TPB) {
        int n = i / K, k = i - n * K;
        float v = 0.0f;
        if (k < Kin)
          v = l ? w_ih[(((l - 1) * 2 + d) * 32 + n) * 16 + k]
                : w_ih0[(d * 32 + n) * 2 + k];
        else if (k < Kin + 8)
          v = l ? w_hh[(((l - 1) * 2 + d) * 32 + n) * 8 + (k - Kin)]
                : w_hh0[(d * 32 + n) * 8 + (k - Kin)];
        W[i] = v;
      }
      for (int i = tid; i < 32; i += TPB)
        sB[(l * 2 + d) * 32 + i] =
            l ? (b_ih[((l - 1) * 2 + d) * 32 + i] + b_hh[((l - 1) * 2 + d) * 32 + i])
              : (b_ih0[d * 32 + i] + b_hh0[d * 32 + i]);
    }
  }
  for (int i = tid; i < 1024; i += TPB) sM[i] = Mws[i];
  for (int i = tid; i < 16; i += TPB) sMb[i] = Mws[1024 + i];
  __syncthreads();

  const int w = tid >> 5, ln = tid & 31, lo = ln & 15, hi = ln >> 4;
  const int b0 = (blockIdx.x * WAVES + w) * 16;

  float (*ain)[16][24]  = sAin[w];
  float (*yout)[16][16] = sYout[w];
  float (*cst)[8]       = sC[w];
  float (*gat)[16]      = sG[w];

  // ---- stage x (coalesced b128): ain[t][m][0..1]; zero layer-0 K pad -----
  {
    v4f xv = *(const v4f*)(x + (size_t)b0 * 8 + ln * 4);
#pragma unroll
    for (int e = 0; e < 4; ++e) {
      int idx = ln * 4 + e;
      ain[(idx >> 1) & 3][idx >> 3][idx & 1] = xv[e];
    }
    for (int q = ln; q < 64; q += 32) {
      ain[q >> 4][q & 15][10] = 0.0f;
      ain[q >> 4][q & 15][11] = 0.0f;
    }
  }
  __builtin_amdgcn_wave_barrier();

  // ---- 6 bi-LSTM layers ---------------------------------------------------
  for (int l = 0; l < 6; ++l) {
    for (int d = 0; d < 2; ++d) {
      const size_t hoff = ((size_t)(2 * l + d) * BT + (size_t)b0) * 8;
      int nid = 2 * l + d + 1;
      if (nid < 12) {  // prefetch next direction's initial state
        __builtin_prefetch(h0 + ((size_t)nid * BT + (size_t)b0) * 8, 0, 0);
        __builtin_prefetch(c0 + ((size_t)nid * BT + (size_t)b0) * 8, 0, 0);
      }
      const float* W = &sW[l ? (768 + ((l - 1) * 2 + d) * 768) : d * 384];
      const float* bvec = &sB[(l * 2 + d) * 32];
      if (l == 0)
        lstm_dir<2, 3>(ain, yout, cst, gat, W, bvec, h0 + hoff, c0 + hoff, d, ln, lo, hi);
      else
        lstm_dir<16, 6>(ain, yout, cst, gat, W, bvec, h0 + hoff, c0 + hoff, d, ln, lo, hi);
    }
    __builtin_amdgcn_wave_barrier();
    if (l < 5) {  // next layer input: copy yout -> ain[..][0..15]
      for (int q = ln; q < 256; q += 32) {
        int t = q >> 6, m = (q >> 2) & 15, f = (q & 3) * 4;
        *(v4f*)&ain[t][m][f] = *(const v4f*)&yout[t][m][f];
      }
      __builtin_amdgcn_wave_barrier();
    }
  }

  // ---- fused head: out = tanh( M * sigmoid(y64) + bias ), one 16x64x16 GEMM
  {
    float mb = sMb[lo];
    v8f acc = {mb, mb, mb, mb, mb, mb, mb, mb};
#pragma unroll
    for (int kt = 0; kt < 16; ++kt) {
      int k = kt * 4 + hi * 2;              // k = t*16 + p
      v2f a = *(const v2f*)&yout[k >> 4][lo][k & 15];
      a.x = sigm(a.x);
      a.y = sigm(a.y);
      v2f bm = *(const v2f*)&sM[lo * 64 + k];
      acc = wmma4(a, bm, acc);
    }
    float* op = out + ((size_t)b0 + 8 * hi) * 16 + lo;  // (m = r+8*hi, n = lo)
#pragma unroll
    for (int r = 0; r < 8; ++r) op[r * 16] = tanh_f(acc[r]);
  }
}

// ---------------------------------------------------------------------------
extern "C" void kernel_launch(void* const* d_in, const int* in_sizes, int n_in,
                              void* d_out, int out_size, void* d_ws,
                              size_t ws_size, hipStream_t stream) {
  (void)in_sizes; (void)n_in; (void)out_size; (void)ws_size;
  const float* x      = (const float*)d_in[0];
  const float* h0     = (const float*)d_in[1];
  const float* c0     = (const float*)d_in[2];
  const float* w_ih0  = (const float*)d_in[3];
  const float* w_hh0  = (const float*)d_in[4];
  const float* b_ih0  = (const float*)d_in[5];
  const float* b_hh0  = (const float*)d_in[6];
  const float* w_ih   = (const float*)d_in[7];
  const float* w_hh   = (const float*)d_in[8];
  const float* b_ih   = (const float*)d_in[9];
  const float* b_hh   = (const float*)d_in[10];
  const float* c1w    = (const float*)d_in[11];
  const float* c1b    = (const float*)d_in[12];
  const float* c2w    = (const float*)d_in[13];
  const float* c2b    = (const float*)d_in[14];
  const float* d1w    = (const float*)d_in[15];
  const float* d1b    = (const float*)d_in[16];
  const float* d2w    = (const float*)d_in[17];
  const float* d2b    = (const float*)d_in[18];
  const float* f1w    = (const float*)d_in[19];
  const float* f1b    = (const float*)d_in[20];
  const float* f2w    = (const float*)d_in[21];
  const float* f2b    = (const float*)d_in[22];
  float* Mws = (float*)d_ws;
  float* out = (float*)d_out;

  head_compose_kernel<<<1, 96, 0, stream>>>(c1w, c1b, c2w, c2b, d1w, d1b, d2w,
                                            d2b, f1w, f1b, f2w, f2b, Mws);
  lstm_wmma_kernel<<<dim3(BT / (16 * WAVES)), dim3(TPB), 0, stream>>>(
      x, h0, c0, w_ih0, w_hh0, b_ih0, b_hh0, w_ih, w_hh, b_ih, b_hh, Mws, out);
}